// MultiUniverseToposAttention_89498528514855
// MI455X (gfx1250) — compile-verified
//
#include <hip/hip_runtime.h>
#include <hip/hip_bf16.h>

// ---------------------------------------------------------------------------
// MultiUniverseToposAttention for MI455X (gfx1250, wave32, WMMA + TDM)
//
//   K1 sig_transpose : sigW[d][o] = sigmoid(W[o][d]) for Wq,Wk,Wv,Wo
//   K2 routing       : logits -> softmax -> top2 -> sel mask + w_u weights
//   K3 gemm_wmma x3  : Qp/Kp/Vp = x @ sigW        (V_WMMA_F32_16X16X4_F32)
//   K4 mask_rope     : mask by sel, RoPE, sigmoid; emits K_cache/V_cache
//   K5 attn          : fused truth (clip-mean) + row-normalize + truth@V
//                      K/V chunks streamed via TENSOR_LOAD_TO_LDS (TDM,
//                      double-buffered, s_wait_tensorcnt), truth built
//                      directly in the WMMA A-register layout
//   K6 gemm_wmma     : final = out_scaled @ sigWo  -> d_out
// ---------------------------------------------------------------------------

typedef __attribute__((ext_vector_type(2))) float v2f;
typedef __attribute__((ext_vector_type(8))) float v8f;
typedef __attribute__((ext_vector_type(4))) unsigned v4u;
typedef __attribute__((ext_vector_type(8))) int v8i;
typedef __attribute__((ext_vector_type(4))) int v4i;

#define SEQ    1024
#define DMODEL 256
#define NUMU   8
#define DU     32

__device__ __forceinline__ float sigmoidf_(float x) {
    return 1.0f / (1.0f + __expf(-x));
}

__device__ __forceinline__ v8f wmma_f32_4(v2f a, v2f b, v8f c) {
    // D = A(16x4 f32) * B(4x16 f32) + C(16x16 f32)
    return __builtin_amdgcn_wmma_f32_16x16x4_f32(
        /*neg_a=*/false, a, /*neg_b=*/false, b,
        /*c_mod=*/(short)0, c, /*reuse_a=*/false, /*reuse_b=*/false);
}

// LDS byte-offset of a __shared__ object (AS3 pointers are 32-bit LDS offsets)
typedef __attribute__((address_space(3))) unsigned char lds_byte;
__device__ __forceinline__ unsigned lds_offset_of(void* p) {
    return (unsigned)(uintptr_t)(lds_byte*)p;
}

// --------------------- TDM: 1-D contiguous global -> LDS --------------------
// D# per CDNA5 ISA ch.8: group0 = {count/flags, lds_addr, global_addr, type},
// group1 = {data_size, tensor_dim0/1, tile_dim0..2, tensor_dim0_stride}.
// 1-D tile of `nelem` f32: tile_dim0 = tensor_dim0 = stride = nelem.
// This toolchain exposes the 6-arg builtin: (g0, g1, g2, g3, g4, cpol).
__device__ __forceinline__ void tdm_load_1d(const float* gptr, unsigned lds_off,
                                            unsigned nelem) {
    unsigned long long ga = (unsigned long long)(uintptr_t)gptr;
    v4u g0;
    g0[0] = 1u;                                   // count=1, user mode
    g0[1] = lds_off;                              // lds_addr (bytes)
    g0[2] = (unsigned)ga;                         // global_addr[31:0]
    g0[3] = (unsigned)(ga >> 32) | (2u << 30);    // global_addr[56:32] | type=2
    v8i g1;
    g1[0] = (int)(2u << 16);                      // wg_mask=0, data_size=4B
    g1[1] = (int)((nelem & 0xffffu) << 16);       // tensor_dim0[15:0]
    g1[2] = (int)((nelem >> 16) | (1u << 16));    // tensor_dim0[31:16], tensor_dim1=1
    g1[3] = (int)((nelem & 0xffffu) << 16);       // tensor_dim1 hi=0, tile_dim0
    g1[4] = 0;                                    // tile_dim1=0, tile_dim2=0
    g1[5] = (int)nelem;                           // tensor_dim0_stride[31:0]
    g1[6] = 0;
    g1[7] = 0;
    v4i z4 = {0, 0, 0, 0};
    v8i z8 = {0, 0, 0, 0, 0, 0, 0, 0};
    __builtin_amdgcn_tensor_load_to_lds(g0, g1, z4, z4, z8, 0);
}

// --------------------------- K1: sigmoid + transpose ------------------------
__global__ void sig_transpose_kernel(const float* __restrict__ Wq,
                                     const float* __restrict__ Wk,
                                     const float* __restrict__ Wv,
                                     const float* __restrict__ Wo,
                                     float* __restrict__ out) {
    int id = blockIdx.x * blockDim.x + threadIdx.x;      // 4*65536 threads
    int mat = id >> 16;
    int e   = id & 65535;
    int o   = e >> 8;
    int d   = e & 255;
    const float* W = (mat == 0) ? Wq : (mat == 1) ? Wk : (mat == 2) ? Wv : Wo;
    out[mat * 65536 + d * 256 + o] = sigmoidf_(W[o * 256 + d]);
}

// ------------------------------- K2: routing --------------------------------
__global__ void routing_kernel(const float* __restrict__ x,
                               const float* __restrict__ Wr,
                               float* __restrict__ sel,
                               float* __restrict__ w_u) {
    int s = blockIdx.x * blockDim.x + threadIdx.x;
    if (s >= SEQ) return;
    const float* xr = x + s * DMODEL;
    float lg[NUMU];
#pragma unroll
    for (int u = 0; u < NUMU; ++u) {
        float acc = 0.0f;
        for (int d = 0; d < DMODEL; ++d)
            acc += xr[d] * sigmoidf_(Wr[u * DMODEL + d]);
        lg[u] = acc;
    }
    float mx = lg[0];
#pragma unroll
    for (int u = 1; u < NUMU; ++u) mx = fmaxf(mx, lg[u]);
    float p[NUMU], sum = 0.0f;
#pragma unroll
    for (int u = 0; u < NUMU; ++u) { p[u] = __expf(lg[u] - mx); sum += p[u]; }
    float inv = 1.0f / sum;
#pragma unroll
    for (int u = 0; u < NUMU; ++u) p[u] *= inv;
    int i0 = 0;
#pragma unroll
    for (int u = 1; u < NUMU; ++u) if (p[u] > p[i0]) i0 = u;
    int i1 = (i0 == 0) ? 1 : 0;
#pragma unroll
    for (int u = 0; u < NUMU; ++u) if (u != i0 && p[u] > p[i1]) i1 = u;
    float denom = p[i0] + p[i1] + 1e-9f;
#pragma unroll
    for (int u = 0; u < NUMU; ++u) {
        float isel = (u == i0 || u == i1) ? 1.0f : 0.0f;
        sel[s * NUMU + u] = isel;
        w_u[s * NUMU + u] = (u == i0) ? p[i0] / denom
                          : (u == i1) ? p[i1] / denom : 0.0f;
    }
}

// --------------------- K3/K6: f32 GEMM via WMMA 16x16x4 ---------------------
// One wave owns a 16x64 C strip: 4 accumulators share each A fetch (4x A reuse).
// All operand loads for a K-step are issued before the WMMA chain so the
// scheduler can hide them with partial s_wait_loadcnt instead of full stalls.
// Lane layout per ISA 7.12.2: m/n = lane&15; h = lane>>4 selects K pair/M half.
__global__ __launch_bounds__(256)
void gemm_wmma_kernel(const float* __restrict__ A, const float* __restrict__ B,
                      float* __restrict__ C, int M, int N, int K) {
    int wave = threadIdx.x >> 5;
    int lane = threadIdx.x & 31;
    int n64  = N >> 6;
    int tile = blockIdx.x * 8 + wave;
    if (tile >= (M >> 4) * n64) return;
    int m0 = (tile / n64) << 4;
    int n0 = (tile % n64) << 6;
    int m  = lane & 15;
    int h  = lane >> 4;

    v8f c0 = {}, c1 = {}, c2 = {}, c3 = {};
#pragma unroll 8
    for (int kk = 0; kk < K; kk += 4) {
        if ((kk & 31) == 0 && kk + 32 < K)       // global_prefetch_b8
            __builtin_prefetch(B + (kk + 32) * N + n0 + lane * 2, 0, 1);
        const float* ap  = A + (m0 + m) * K + kk + 2 * h;
        const float* b0p = B + (kk + 2 * h + 0) * N + n0 + m;
        const float* b1p = B + (kk + 2 * h + 1) * N + n0 + m;
        // issue all 10 scalar loads for this K-step up front
        v2f a;  a.x  = ap[0];    a.y  = ap[1];
        v2f b0; b0.x = b0p[0];   b0.y = b1p[0];
        v2f b1; b1.x = b0p[16];  b1.y = b1p[16];
        v2f b2; b2.x = b0p[32];  b2.y = b1p[32];
        v2f b3; b3.x = b0p[48];  b3.y = b1p[48];
        c0 = wmma_f32_4(a, b0, c0);
        c1 = wmma_f32_4(a, b1, c1);
        c2 = wmma_f32_4(a, b2, c2);
        c3 = wmma_f32_4(a, b3, c3);
    }
    float* crow = C + n0 + m;
#pragma unroll
    for (int r = 0; r < 8; ++r) {
        int row = m0 + r + 8 * h;
        crow[row * N +  0] = c0[r];
        crow[row * N + 16] = c1[r];
        crow[row * N + 32] = c2[r];
        crow[row * N + 48] = c3[r];
    }
}

// ------------------------- K4: mask + RoPE + sigmoid ------------------------
__global__ void mask_rope_kernel(const float* __restrict__ Qp,
                                 const float* __restrict__ Kp,
                                 const float* __restrict__ Vp,
                                 const float* __restrict__ fcos,
                                 const float* __restrict__ fsin,
                                 const float* __restrict__ sel,
                                 float* __restrict__ Qs,
                                 float* __restrict__ Ks,
                                 float* __restrict__ Vt,
                                 float* __restrict__ kcache,
                                 float* __restrict__ vcache) {
    int id = blockIdx.x * blockDim.x + threadIdx.x;   // SEQ*NUMU*16 threads
    int s   = id >> 7;
    int rem = id & 127;
    int u   = rem >> 4;
    int d2  = rem & 15;

    float sv = sel[s * NUMU + u];
    float cv = fcos[s * 16 + d2];
    float sn = fsin[s * 16 + d2];
    int base = s * DMODEL + u * DU + 2 * d2;

    float qre = Qp[base] * sv, qim = Qp[base + 1] * sv;
    float kre = Kp[base] * sv, kim = Kp[base + 1] * sv;
    float v0  = Vp[base] * sv, v1  = Vp[base + 1] * sv;

    float qr = qre * cv - qim * sn, qi = qre * sn + qim * cv;
    float kr = kre * cv - kim * sn, ki = kre * sn + kim * cv;

    kcache[base]     = kr;  kcache[base + 1] = ki;
    vcache[base]     = v0;  vcache[base + 1] = v1;

    int tbase = u * (SEQ * DU) + s * DU + 2 * d2;
    Qs[tbase]     = sigmoidf_(qr);  Qs[tbase + 1] = sigmoidf_(qi);
    Ks[tbase]     = sigmoidf_(kr);  Ks[tbase + 1] = sigmoidf_(ki);
    Vt[tbase]     = v0;             Vt[tbase + 1] = v1;
}

// ----------------- K5: fused truth + normalize + truth @ V ------------------
// Block = 4 waves sharing one universe; each wave owns a 16-row s-tile.
// K/V chunks arrive via TDM (wave 0 issues; double-buffered; TENSORcnt).
__global__ __launch_bounds__(128)
void attn_kernel(const float* __restrict__ Qs, const float* __restrict__ Ks,
                 const float* __restrict__ Vt, const float* __restrict__ w_u,
                 float* __restrict__ outs) {
    __shared__ float ldsK[2][16 * DU];
    __shared__ float ldsV[2][16 * DU];

    int wave = threadIdx.x >> 5;
    int lane = threadIdx.x & 31;
    int u    = blockIdx.x >> 4;                       // 128 blocks: u = 0..7
    int s0   = (((blockIdx.x & 15) << 2) + wave) << 4;
    int m    = lane & 15;
    int h    = lane >> 4;

    const float* Ku = Ks + u * (SEQ * DU);
    const float* Vu = Vt + u * (SEQ * DU);

    // preload chunk 0 via TDM (EXEC-independent, per-wave TENSORcnt)
    if (wave == 0) {
        tdm_load_1d(Ku, lds_offset_of(&ldsK[0][0]), 16 * DU);
        tdm_load_1d(Vu, lds_offset_of(&ldsV[0][0]), 16 * DU);
    }

    const float* qrow = Qs + u * (SEQ * DU) + (s0 + m) * DU;
    float q[DU];
#pragma unroll
    for (int d = 0; d < DU; ++d) q[d] = qrow[d];

    v8f c0 = {}, c1 = {};
    float rowsum = 0.0f;

    for (int tc = 0; tc < SEQ; tc += 16) {
        int cur = (tc >> 4) & 1;
        if (wave == 0)
            __builtin_amdgcn_s_wait_tensorcnt(0);     // current buffers landed
        __syncthreads();                              // publish LDS to all waves
        if (wave == 0 && tc + 16 < SEQ) {             // overlap next chunk DMA
            tdm_load_1d(Ku + (tc + 16) * DU,
                        lds_offset_of(&ldsK[cur ^ 1][0]), 16 * DU);
            tdm_load_1d(Vu + (tc + 16) * DU,
                        lds_offset_of(&ldsV[cur ^ 1][0]), 16 * DU);
        }
        const float* kc = ldsK[cur];
        const float* vc = ldsV[cur];

        // truth values this lane contributes to the A matrix (8 of 16 t's)
        float t8[8];
#pragma unroll
        for (int kk = 0; kk < 4; ++kk) {
#pragma unroll
            for (int j = 0; j < 2; ++j) {
                const float* krow = kc + (4 * kk + 2 * h + j) * DU;
                float acc = 0.0f;
#pragma unroll
                for (int d = 0; d < DU; ++d) {
                    float val = 1.0f - q[d] + krow[d];
                    val = fminf(fmaxf(val, 0.0f), 1.0f);
                    acc += val;
                }
                t8[2 * kk + j] = acc;
                rowsum += acc;
            }
        }
        // out(16x32) += truth(16x16) @ V(16x32) : 2 d-halves x 4 K-steps
#pragma unroll
        for (int kk = 0; kk < 4; ++kk) {
            v2f a; a.x = t8[2 * kk]; a.y = t8[2 * kk + 1];
            const float* v0p = vc + (4 * kk + 2 * h + 0) * DU;
            const float* v1p = vc + (4 * kk + 2 * h + 1) * DU;
            v2f b0; b0.x = v0p[m];      b0.y = v1p[m];
            v2f b1; b1.x = v0p[16 + m]; b1.y = v1p[16 + m];
            c0 = wmma_f32_4(a, b0, c0);
            c1 = wmma_f32_4(a, b1, c1);
        }
    }

    // full raw row sum for row m lives split across lane pairs (l, l^16)
    float rs = rowsum + __shfl_xor(rowsum, 16, 32);

#pragma unroll
    for (int r = 0; r < 8; ++r) {
        int row = r + 8 * h;                       // C-layout row for this lane
        float denom  = __shfl(rs, row, 32) + 32.0f * 1e-9f;
        float wscale = w_u[(s0 + row) * NUMU + u];
        float scale  = wscale / denom;
        outs[(s0 + row) * DMODEL + u * DU + m]      = c0[r] * scale;
        outs[(s0 + row) * DMODEL + u * DU + 16 + m] = c1[r] * scale;
    }
}

// ------------------------------- launcher -----------------------------------
extern "C" void kernel_launch(void* const* d_in, const int* in_sizes, int n_in,
                              void* d_out, int out_size, void* d_ws, size_t ws_size,
                              hipStream_t stream) {
    const float* x    = (const float*)d_in[0];   // (1,1024,256)
    const float* fcos = (const float*)d_in[1];   // (1024,16)
    const float* fsin = (const float*)d_in[2];   // (1024,16)
    const float* Wr   = (const float*)d_in[3];   // (8,256)
    const float* Wq   = (const float*)d_in[4];   // (256,256)
    const float* Wk   = (const float*)d_in[5];
    const float* Wv   = (const float*)d_in[6];
    const float* Wo   = (const float*)d_in[7];

    float* out    = (float*)d_out;
    float* final_ = out;                          // 1024*256
    float* kcache = out + SEQ * DMODEL;           // 1024*256
    float* vcache = out + 2 * SEQ * DMODEL;       // 1024*256

    float* ws = (float*)d_ws;
    float* sigW  = ws;                     // 4 * 65536 (q,k,v,o)
    float* sigWq = sigW;
    float* sigWk = sigW + 65536;
    float* sigWv = sigW + 131072;
    float* sigWo = sigW + 196608;
    float* Qp   = ws + 262144;             // 65536 each
    float* Kp   = ws + 327680;
    float* Vp   = ws + 393216;
    float* Qs   = ws + 458752;             // [u][s][32]
    float* Ksb  = ws + 524288;
    float* Vtb  = ws + 589824;
    float* outs = ws + 655360;             // out_scaled [s][256]
    float* sel  = ws + 720896;             // [s][8]
    float* wu   = ws + 729088;             // [s][8]

    sig_transpose_kernel<<<(4 * 65536) / 256, 256, 0, stream>>>(Wq, Wk, Wv, Wo, sigW);
    routing_kernel<<<SEQ / 256, 256, 0, stream>>>(x, Wr, sel, wu);

    const int gemm_blocks = (SEQ / 16) * (DMODEL / 64) / 8;   // 32
    gemm_wmma_kernel<<<gemm_blocks, 256, 0, stream>>>(x, sigWq, Qp, SEQ, DMODEL, DMODEL);
    gemm_wmma_kernel<<<gemm_blocks, 256, 0, stream>>>(x, sigWk, Kp, SEQ, DMODEL, DMODEL);
    gemm_wmma_kernel<<<gemm_blocks, 256, 0, stream>>>(x, sigWv, Vp, SEQ, DMODEL, DMODEL);

    mask_rope_kernel<<<(SEQ * NUMU * 16) / 256, 256, 0, stream>>>(
        Qp, Kp, Vp, fcos, fsin, sel, Qs, Ksb, Vtb, kcache, vcache);

    attn_kernel<<<NUMU * (SEQ / 16) / 4, 128, 0, stream>>>(Qs, Ksb, Vtb, wu, outs);

    gemm_wmma_kernel<<<gemm_blocks, 256, 0, stream>>>(outs, sigWo, final_, SEQ, DMODEL, DMODEL);
}